// dD_predictor_52553219834470
// MI455X (gfx1250) — compile-verified
//
#include <hip/hip_runtime.h>
#include <hip/hip_bf16.h>

typedef __attribute__((ext_vector_type(2))) float v2f;
typedef __attribute__((ext_vector_type(8))) float v8f;

#define HDIM 128
#define NGRAPH 512
#define NEG_SLOPE 0.2f
#define KPAD 16            // F_IN=10 zero-padded to 16

// ---------------------------------------------------------------------------
// GEMM:  out[M x 128] = act( inA (+ inB) ) @ W[K x 128]  (+ bias)
// K is compile-time (multiple of 4, no guards).  MODE: 0 plain, 1 residual,
// 2 relu(input).  One wave per 16-row strip, 8 col tiles, WMMA f32 16x16x4.
// W staged in LDS (K*128*4 bytes <= 64 KB).  blockDim.x = 128 (4 waves).
// ---------------------------------------------------------------------------
template <int K, int MODE>
__global__ void gemm_wmma_k(const float* __restrict__ inA,
                            const float* __restrict__ inB,    // residual (MODE 1)
                            const float* __restrict__ W,      // [K x 128]
                            const float* __restrict__ bias,   // [128] or null
                            float* __restrict__ out,          // [M x 128]
                            int M)
{
    __shared__ float wlds[K * HDIM];
    // cooperative W stage: all threads participate before any wave may exit
    for (int i = threadIdx.x; i < K * HDIM / 4; i += 128)
        ((float4*)wlds)[i] = ((const float4*)W)[i];
    __syncthreads();

    const int lane  = threadIdx.x & 31;
    const int wave  = threadIdx.x >> 5;
    const int strip = blockIdx.x * 4 + wave;
    const int row0  = strip * 16;
    if (row0 >= M) return;                 // wave-uniform: EXEC stays all-1s

    const int half = lane >> 4;            // 0: lanes 0-15, 1: lanes 16-31
    const int l16  = lane & 15;

    v8f acc[8] = {};

    const float2* arow  = (const float2*)(inA + (size_t)(row0 + l16) * K);
    const float2* arowR = (MODE == 1) ? (const float2*)(inB + (size_t)(row0 + l16) * K)
                                      : nullptr;

#pragma unroll 4
    for (int k0 = 0; k0 < K; k0 += 4) {
        const int ka = k0 + 2 * half;      // even -> aligned float2
        float2 av = arow[ka >> 1];
        if (MODE == 1) {
            float2 rv = arowR[ka >> 1];
            av.x += rv.x; av.y += rv.y;
        } else if (MODE == 2) {
            av.x = fmaxf(av.x, 0.f); av.y = fmaxf(av.y, 0.f);
        }
        v2f afrag; afrag[0] = av.x; afrag[1] = av.y;

        const float* w0 = &wlds[(ka    ) * HDIM + l16];
        const float* w1 = &wlds[(ka + 1) * HDIM + l16];
#pragma unroll
        for (int t = 0; t < 8; ++t) {
            v2f bfrag;
            bfrag[0] = w0[t * 16];
            bfrag[1] = w1[t * 16];
            acc[t] = __builtin_amdgcn_wmma_f32_16x16x4_f32(
                false, afrag, false, bfrag, (short)0, acc[t], false, false);
        }
    }

#pragma unroll
    for (int t = 0; t < 8; ++t) {
        const int col = t * 16 + l16;
        const float bv = bias ? bias[col] : 0.f;
#pragma unroll
        for (int v = 0; v < 8; ++v) {
            const int r = row0 + v + 8 * half;     // C layout: vgpr v -> rows v, v+8
            out[(size_t)r * HDIM + col] = acc[t][v] + bv;
        }
    }
}

// ---------------------------------------------------------------------------
// Zero-pad helpers (K=10 -> 16)
// ---------------------------------------------------------------------------
__global__ void pad_x_kernel(const float* __restrict__ x, float* __restrict__ xpad,
                             int N, int Fin)
{
    size_t idx = (size_t)blockIdx.x * blockDim.x + threadIdx.x;
    if (idx >= (size_t)N * KPAD) return;
    int row = (int)(idx >> 4), col = (int)(idx & (KPAD - 1));
    xpad[idx] = (col < Fin) ? x[(size_t)row * Fin + col] : 0.f;
}

__global__ void pad_w_kernel(const float* __restrict__ w, float* __restrict__ wpad, int Fin)
{
    int idx = blockIdx.x * blockDim.x + threadIdx.x;
    if (idx >= KPAD * HDIM) return;
    int row = idx >> 7;
    wpad[idx] = (row < Fin) ? w[idx] : 0.f;
}

// ---------------------------------------------------------------------------
// alpha_s[i] = h[i,:] . a_src ; alpha_d[i] = h[i,:] . a_dst
// ---------------------------------------------------------------------------
__global__ void gat_alpha(const float* __restrict__ h,
                          const float* __restrict__ a_s,
                          const float* __restrict__ a_d,
                          float* __restrict__ alpha_s,
                          float* __restrict__ alpha_d, int N)
{
    int i = blockIdx.x * blockDim.x + threadIdx.x;
    if (i >= N) return;
    const float4* hp = (const float4*)(h + (size_t)i * HDIM);
    const float4* sp = (const float4*)a_s;
    const float4* dp = (const float4*)a_d;
    float s = 0.f, d = 0.f;
#pragma unroll 8
    for (int j = 0; j < HDIM / 4; ++j) {
        float4 hv = hp[j], sv = sp[j], dv = dp[j];
        s += hv.x * sv.x + hv.y * sv.y + hv.z * sv.z + hv.w * sv.w;
        d += hv.x * dv.x + hv.y * dv.y + hv.z * dv.z + hv.w * dv.w;
    }
    alpha_s[i] = s; alpha_d[i] = d;
}

// ---------------------------------------------------------------------------
// Per-layer init: out rows <- bias, segment state <- identity
// ---------------------------------------------------------------------------
__global__ void gat_init_layer(float* __restrict__ out, const float* __restrict__ bias,
                               unsigned* __restrict__ keymax, float* __restrict__ ssum,
                               int N)
{
    size_t idx = (size_t)blockIdx.x * blockDim.x + threadIdx.x;
    if (idx < (size_t)N * HDIM) out[idx] = bias[idx & (HDIM - 1)];
    if (idx < (size_t)N) { keymax[idx] = 0u; ssum[idx] = 0.f; }
}

__device__ __forceinline__ unsigned f2key(float f) {
    unsigned b = __float_as_uint(f);
    return (b & 0x80000000u) ? ~b : (b | 0x80000000u);
}
__device__ __forceinline__ float key2f(unsigned k) {
    return (k & 0x80000000u) ? __uint_as_float(k & 0x7fffffffu) : __uint_as_float(~k);
}

// ---------------------------------------------------------------------------
// Pass 1: segment max of leaky_relu(alpha_s[src]+alpha_d[dst]) over dst
// ---------------------------------------------------------------------------
__global__ void gat_edge_max(const int* __restrict__ src, const int* __restrict__ dst,
                             const float* __restrict__ as_, const float* __restrict__ ad_,
                             unsigned* __restrict__ keymax, int E, int N)
{
    int e = blockIdx.x * blockDim.x + threadIdx.x;
    if (e >= E + N) return;
    int s = (e < E) ? src[e] : (e - E);
    int d = (e < E) ? dst[e] : (e - E);
    float l = as_[s] + ad_[d];
    l = (l >= 0.f) ? l : NEG_SLOPE * l;
    atomicMax(&keymax[d], f2key(l));
}

// ---------------------------------------------------------------------------
// Pass 2: e = exp(logit - max[dst]); store e; ssum[dst] += e
// ---------------------------------------------------------------------------
__global__ void gat_edge_exp(const int* __restrict__ src, const int* __restrict__ dst,
                             const float* __restrict__ as_, const float* __restrict__ ad_,
                             const unsigned* __restrict__ keymax,
                             float* __restrict__ ssum, float* __restrict__ ev,
                             int E, int N)
{
    int e = blockIdx.x * blockDim.x + threadIdx.x;
    if (e >= E + N) return;
    int s = (e < E) ? src[e] : (e - E);
    int d = (e < E) ? dst[e] : (e - E);
    float l = as_[s] + ad_[d];
    l = (l >= 0.f) ? l : NEG_SLOPE * l;
    float x = __expf(l - key2f(keymax[d]));
    ev[e] = x;
    atomicAdd(&ssum[d], x);
}

// ---------------------------------------------------------------------------
// Pass 3: out[dst,:] += (ev/ssum[dst]) * h[src,:].  One wave per edge;
// float4 gather of h[src] + 4 f32 atomics per lane (L2-resident traffic).
// ---------------------------------------------------------------------------
__global__ void gat_edge_scatter(const int* __restrict__ src, const int* __restrict__ dst,
                                 const float* __restrict__ ev, const float* __restrict__ ssum,
                                 const float* __restrict__ h, float* __restrict__ out,
                                 int E, int N)
{
    int lane = threadIdx.x & 31;
    long e = (long)blockIdx.x * 8 + (threadIdx.x >> 5);    // blockDim = 256 = 8 waves
    if (e >= (long)E + N) return;
    int s = (e < E) ? src[e] : (int)(e - E);
    int d = (e < E) ? dst[e] : (int)(e - E);
    float w = ev[e] / ssum[d];
    float4 hv = ((const float4*)(h + (size_t)s * HDIM))[lane];
    float* op = out + (size_t)d * HDIM + lane * 4;
    atomicAdd(op + 0, hv.x * w);
    atomicAdd(op + 1, hv.y * w);
    atomicAdd(op + 2, hv.z * w);
    atomicAdd(op + 3, hv.w * w);
}

// ---------------------------------------------------------------------------
// Pooling
// ---------------------------------------------------------------------------
__global__ void pool_init(float* __restrict__ psum, float* __restrict__ cnts)
{
    int idx = blockIdx.x * blockDim.x + threadIdx.x;
    if (idx < NGRAPH * HDIM) psum[idx] = 0.f;
    if (idx < NGRAPH) cnts[idx] = 0.f;
}

__global__ void pool_accum(const float* __restrict__ x3, const int* __restrict__ batch,
                           float* __restrict__ psum, float* __restrict__ cnts, int N)
{
    int i = blockIdx.x;            // one 128-thread block per node
    int c = threadIdx.x;
    int b = batch[i];
    atomicAdd(&psum[(size_t)b * HDIM + c], x3[(size_t)i * HDIM + c]);
    if (c == 0) atomicAdd(&cnts[b], 1.f);
}

__global__ void pool_finalize(const float* __restrict__ psum, const float* __restrict__ cnts,
                              float* __restrict__ pooled)
{
    int idx = blockIdx.x * blockDim.x + threadIdx.x;
    if (idx >= NGRAPH * HDIM) return;
    pooled[idx] = psum[idx] / fmaxf(cnts[idx >> 7], 1.f);
}

// ---------------------------------------------------------------------------
// Head: y[g] = (relu(h2[g,:]) . fc2_w + fc2_b) / 100 ; one wave per graph
// ---------------------------------------------------------------------------
__global__ void head_kernel(const float* __restrict__ h2, const float* __restrict__ fc2w,
                            const float* __restrict__ fc2b, float* __restrict__ y)
{
    int g = blockIdx.x, lane = threadIdx.x;
    float4 hv = ((const float4*)(h2 + (size_t)g * HDIM))[lane];
    float4 wv = ((const float4*)fc2w)[lane];
    float p = fmaxf(hv.x, 0.f) * wv.x + fmaxf(hv.y, 0.f) * wv.y +
              fmaxf(hv.z, 0.f) * wv.z + fmaxf(hv.w, 0.f) * wv.w;
#pragma unroll
    for (int o = 16; o > 0; o >>= 1) p += __shfl_down(p, o);
    if (lane == 0) y[g] = (p + fc2b[0]) * 0.01f;
}

// ---------------------------------------------------------------------------
extern "C" void kernel_launch(void* const* d_in, const int* in_sizes, int n_in,
                              void* d_out, int out_size, void* d_ws, size_t ws_size,
                              hipStream_t stream)
{
    const float* x     = (const float*)d_in[0];
    const int*   eidx  = (const int*)d_in[1];
    const int*   batch = (const int*)d_in[3];     // d_in[2] = edge_attr (unused)

    const int N    = in_sizes[3];                 // 50000
    const int E    = in_sizes[1] / 2;             // 1600000
    const int F_IN = in_sizes[0] / N;             // 10
    const int* srcp = eidx;
    const int* dstp = eidx + E;

    const float *W1 = (const float*)d_in[4],  *as1 = (const float*)d_in[5],
                *ad1 = (const float*)d_in[6], *b1 = (const float*)d_in[7];
    const float *W2 = (const float*)d_in[8],  *as2 = (const float*)d_in[9],
                *ad2 = (const float*)d_in[10], *b2 = (const float*)d_in[11];
    const float *W3 = (const float*)d_in[12], *as3 = (const float*)d_in[13],
                *ad3 = (const float*)d_in[14], *b3 = (const float*)d_in[15];
    const float *l1w = (const float*)d_in[16], *l1b = (const float*)d_in[17];
    const float *f1w = (const float*)d_in[18], *f1b = (const float*)d_in[19];
    const float *f2w = (const float*)d_in[20], *f2b = (const float*)d_in[21];

    // ---- workspace carve-up (~90 MB) ----
    const size_t NH = (size_t)N * HDIM;
    float* ws = (float*)d_ws;
    float*    h_    = ws;                 ws += NH;   // transformed feats (per layer)
    float*    x1    = ws;                 ws += NH;
    float*    x2    = ws;                 ws += NH;
    float*    x3    = x1;                 // x1 dead after layer-3 GEMM: alias
    float*    as_   = ws;                 ws += N;
    float*    ad_   = ws;                 ws += N;
    unsigned* keym  = (unsigned*)ws;      ws += N;
    float*    ssum  = ws;                 ws += N;
    float*    ev    = ws;                 ws += (size_t)E + N;
    float*    psum  = ws;                 ws += (size_t)NGRAPH * HDIM;
    float*    cnts  = ws;                 ws += NGRAPH;
    float*    pooled= ws;                 ws += (size_t)NGRAPH * HDIM;
    float*    mh1   = ws;                 ws += (size_t)NGRAPH * HDIM;
    float*    mh2   = ws;                 ws += (size_t)NGRAPH * HDIM;
    float*    xpad  = ws;                 ws += (size_t)N * KPAD;
    float*    w1pad = ws;                 ws += (size_t)KPAD * HDIM;
    (void)ws_size; (void)n_in; (void)out_size;

    const int EN = E + N;
    const int gemmBlocksN = (N / 16 + 3) / 4;     // 4 waves/block, 16 rows/wave
    const int gemmBlocksG = (NGRAPH / 16 + 3) / 4;
    const int edgeBlocks  = (EN + 255) / 256;
    const int initBlocks  = (int)((NH + 255) / 256);

    // pad x (K=10 -> 16) and W1 once per launch
    pad_x_kernel<<<(int)(((size_t)N * KPAD + 255) / 256), 256, 0, stream>>>(x, xpad, N, F_IN);
    pad_w_kernel<<<(KPAD * HDIM + 255) / 256, 256, 0, stream>>>(W1, w1pad, F_IN);

    auto gat_edges = [&](const float* avs, const float* avd, const float* bias, float* xout) {
        gat_alpha<<<(N + 255) / 256, 256, 0, stream>>>(h_, avs, avd, as_, ad_, N);
        gat_init_layer<<<initBlocks, 256, 0, stream>>>(xout, bias, keym, ssum, N);
        gat_edge_max<<<edgeBlocks, 256, 0, stream>>>(srcp, dstp, as_, ad_, keym, E, N);
        gat_edge_exp<<<edgeBlocks, 256, 0, stream>>>(srcp, dstp, as_, ad_, keym, ssum, ev, E, N);
        gat_edge_scatter<<<(EN + 7) / 8, 256, 0, stream>>>(srcp, dstp, ev, ssum, h_, xout, E, N);
    };

    // layer 1: h = xpad @ W1pad
    gemm_wmma_k<KPAD, 0><<<gemmBlocksN, 128, 0, stream>>>(xpad, nullptr, w1pad, nullptr, h_, N);
    gat_edges(as1, ad1, b1, x1);
    // layer 2: h = x1 @ W2
    gemm_wmma_k<HDIM, 0><<<gemmBlocksN, 128, 0, stream>>>(x1, nullptr, W2, nullptr, h_, N);
    gat_edges(as2, ad2, b2, x2);
    // layer 3: h = (x2 + x1) @ W3    (x3 aliases x1; x1 only read here)
    gemm_wmma_k<HDIM, 1><<<gemmBlocksN, 128, 0, stream>>>(x2, x1, W3, nullptr, h_, N);
    gat_edges(as3, ad3, b3, x3);

    // global mean pool + MLP head
    pool_init<<<(NGRAPH * HDIM + 255) / 256, 256, 0, stream>>>(psum, cnts);
    pool_accum<<<N, HDIM, 0, stream>>>(x3, batch, psum, cnts, N);
    pool_finalize<<<(NGRAPH * HDIM + 255) / 256, 256, 0, stream>>>(psum, cnts, pooled);

    gemm_wmma_k<HDIM, 2><<<gemmBlocksG, 128, 0, stream>>>(pooled, nullptr, l1w, l1b, mh1, NGRAPH);
    gemm_wmma_k<HDIM, 2><<<gemmBlocksG, 128, 0, stream>>>(mh1,    nullptr, f1w, f1b, mh2, NGRAPH);
    head_kernel<<<NGRAPH, 32, 0, stream>>>(mh2, f2w, f2b, (float*)d_out);
}